// DiscriminativeLoss_13159779795024
// MI455X (gfx1250) — compile-verified
//
#include <hip/hip_runtime.h>
#include <math.h>

#define Bsz 16
#define Nn  4096
#define Kk  64
#define Ee  32

#define S1  8     // k1 N-slices  -> 128 blocks
#define S2  16    // k2 N-slices  -> 256 blocks
#define CH  64    // staged chunk rows

#define MSTR  72  // masks tile LDS row stride (floats): 2*72 % 64 == 16 -> disjoint A windows
#define ESTR1 40  // emb tile stride in k1 (B operand): 2*40 % 64 == 16 -> disjoint
#define ESTR2 36  // emb tile stride in k2 (A operand): banks = mult-of-4, +2 shift disjoint
#define CSTR  36  // centroid LDS stride in k2 (B operand)

#define DELTA_D 1.5f
#define DELTA_V 0.5f
#define ALPHA   1.0f
#define BETA    1.0f
#define GAMMA   0.001f

typedef __attribute__((ext_vector_type(2))) float v2f;
typedef __attribute__((ext_vector_type(8))) float v8f;
typedef __attribute__((ext_vector_type(4))) int   v4i;

#ifndef __has_builtin
#define __has_builtin(x) 0
#endif
#if __has_builtin(__builtin_amdgcn_global_load_async_to_lds_b128)
#define HAVE_ASYNC 1
#else
#define HAVE_ASYNC 0
#endif
#if __has_builtin(__builtin_amdgcn_s_wait_asynccnt)
#define HAVE_WAITA 1
#else
#define HAVE_WAITA 0
#endif

// 16-byte global -> LDS copy (async on CDNA5, sync float4 fallback).
__device__ __forceinline__ void cp16_to_lds(const float* g, float* l) {
#if HAVE_ASYNC
    __builtin_amdgcn_global_load_async_to_lds_b128(
        (__attribute__((address_space(1))) v4i*)(v4i*)g,
        (__attribute__((address_space(3))) v4i*)(v4i*)l, 0, 0);
#else
    *(float4*)l = *(const float4*)g;
#endif
}
__device__ __forceinline__ void wait_async_6() {   // allow next chunk's 6 ops in flight
#if HAVE_ASYNC
#if HAVE_WAITA
    __builtin_amdgcn_s_wait_asynccnt(6);
#else
    asm volatile("s_wait_asynccnt 0x6" ::: "memory");
#endif
#endif
}
__device__ __forceinline__ void wait_async_0() {
#if HAVE_ASYNC
#if HAVE_WAITA
    __builtin_amdgcn_s_wait_asynccnt(0);
#else
    asm volatile("s_wait_asynccnt 0x0" ::: "memory");
#endif
#endif
}

// ---------------------------------------------------------------------------
// Kernel 1: per-(batch, N-slice) partial sums = m^T * emb  and partial counts.
// 8 waves: wave w owns tile (kt=w>>1, et=w&1) of the 64x32 sums matrix.
// Double-buffered async LDS staging of 64-row chunks (6 b128 ops/wave/chunk).
// ---------------------------------------------------------------------------
__global__ __launch_bounds__(256) void k1_partial(
    const float* __restrict__ emb, const float* __restrict__ masks,
    float* __restrict__ psum /* [B*S1][K*E] */, float* __restrict__ pcnt /* [B*S1][K] */)
{
    const int b  = blockIdx.x / S1;
    const int s  = blockIdx.x % S1;
    const int t  = threadIdx.x;
    const int n0 = s * (Nn / S1);
    const float* Mg = masks + ((size_t)b * Nn + n0) * Kk;
    const float* Eg = emb   + ((size_t)b * Nn + n0) * Ee;

    __shared__ __align__(16) float Mt[2][CH * MSTR];
    __shared__ __align__(16) float Et[2][CH * ESTR1];
    __shared__ float scnt[256];

    const int NCH = (Nn / S1) / CH;   // 8 chunks

    auto stage = [&](int c) {
        const float* mg = Mg + (size_t)c * CH * Kk;
        const float* eg = Eg + (size_t)c * CH * Ee;
        float* mt = Mt[c & 1];
        float* et = Et[c & 1];
#pragma unroll
        for (int i = 0; i < 4; ++i) {                  // masks: 16KB
            const int f = (i * 256 + t) * 4;
            cp16_to_lds(mg + f, &mt[(f >> 6) * MSTR + (f & 63)]);
        }
#pragma unroll
        for (int i = 0; i < 2; ++i) {                  // emb: 8KB
            const int f = (i * 256 + t) * 4;
            cp16_to_lds(eg + f, &et[(f >> 5) * ESTR1 + (f & 31)]);
        }
    };

    const int w = t >> 5, lane = t & 31, half = lane >> 4, l15 = lane & 15;
    const int kbase = (w >> 1) * 16, ebase = (w & 1) * 16;

    stage(0);
    float cntp = 0.f;
    v8f acc = {};

    for (int c = 0; c < NCH; ++c) {
        if (c + 1 < NCH) { stage(c + 1); wait_async_6(); } else { wait_async_0(); }
        __syncthreads();
        const float* mt = Mt[c & 1];
        const float* et = Et[c & 1];
        {   // counts partial over this chunk
            const int k = t & 63, q = t >> 6;
            for (int nl = q; nl < CH; nl += 4) cntp += mt[nl * MSTR + k];
        }
        for (int nl = 0; nl < CH; nl += 4) {
            const int r = nl + 2 * half;
            v2f a, bb;
            a[0]  = mt[r * MSTR + kbase + l15];         // A: M=k rows, wmma-K=n
            a[1]  = mt[(r + 1) * MSTR + kbase + l15];
            bb[0] = et[r * ESTR1 + ebase + l15];        // B: rows=n, cols=e
            bb[1] = et[(r + 1) * ESTR1 + ebase + l15];
            acc = __builtin_amdgcn_wmma_f32_16x16x4_f32(
                      false, a, false, bb, (short)0, acc, false, false);
        }
        __syncthreads();
    }

    scnt[t] = cntp;
    __syncthreads();
    if (t < Kk)
        pcnt[(size_t)blockIdx.x * Kk + t] =
            scnt[t] + scnt[t + 64] + scnt[t + 128] + scnt[t + 192];

    float* ps = psum + (size_t)blockIdx.x * (Kk * Ee);
#pragma unroll
    for (int r = 0; r < 8; ++r)
        ps[(kbase + r + 8 * half) * Ee + ebase + l15] = acc[r];
}

// ---------------------------------------------------------------------------
// Kernel 1b: fixed-order combine of slice partials -> centroids.
// ---------------------------------------------------------------------------
__global__ __launch_bounds__(256) void k1b_combine(
    const float* __restrict__ psum, const float* __restrict__ pcnt,
    const int* __restrict__ size, float* __restrict__ cent)
{
    const int b = blockIdx.x, t = threadIdx.x;
    __shared__ float cnt[Kk];
    if (t < Kk) {
        float c = 0.f;
        for (int s = 0; s < S1; ++s) c += pcnt[(size_t)(b * S1 + s) * Kk + t];
        cnt[t] = c;
    }
    __syncthreads();
    const int sz = size[b];
    for (int i = t; i < Kk * Ee; i += 256) {
        const int k = i >> 5;   // Ee == 32
        float sm = 0.f;
        for (int s = 0; s < S1; ++s) sm += psum[(size_t)(b * S1 + s) * (Kk * Ee) + i];
        cent[(size_t)b * Kk * Ee + i] = (k < sz) ? sm / fmaxf(cnt[k], 1.f) : 0.f;
    }
}

// ---------------------------------------------------------------------------
// Kernel 2: per-(batch, N-slice) L_v partials. WMMA Gram tiles + fused hinge.
// ---------------------------------------------------------------------------
__global__ __launch_bounds__(256) void k2_partial(
    const float* __restrict__ emb, const float* __restrict__ masks,
    const int* __restrict__ size, const float* __restrict__ cent,
    float* __restrict__ pnum, float* __restrict__ pden)
{
    const int b  = blockIdx.x / S2;
    const int s  = blockIdx.x % S2;
    const int t  = threadIdx.x;
    const int n0 = s * (Nn / S2);
    const float* Mg = masks + ((size_t)b * Nn + n0) * Kk;
    const float* Eg = emb   + ((size_t)b * Nn + n0) * Ee;
    const float* Cb = cent  + (size_t)b * Kk * Ee;
    const int sz = size[b];

    __shared__ __align__(16) float Mt[2][CH * MSTR];
    __shared__ __align__(16) float Et[2][CH * ESTR2];
    __shared__ float cl[Kk * CSTR];
    __shared__ float cn2[Kk];
    __shared__ float en2[CH];
    __shared__ float red[512];

    const int NCH = (Nn / S2) / CH;   // 4 chunks

    auto stage = [&](int c) {
        const float* mg = Mg + (size_t)c * CH * Kk;
        const float* eg = Eg + (size_t)c * CH * Ee;
        float* mt = Mt[c & 1];
        float* et = Et[c & 1];
#pragma unroll
        for (int i = 0; i < 4; ++i) {
            const int f = (i * 256 + t) * 4;
            cp16_to_lds(mg + f, &mt[(f >> 6) * MSTR + (f & 63)]);
        }
#pragma unroll
        for (int i = 0; i < 2; ++i) {
            const int f = (i * 256 + t) * 4;
            cp16_to_lds(eg + f, &et[(f >> 5) * ESTR2 + (f & 31)]);
        }
    };

    for (int i = t; i < Kk * Ee; i += 256)             // centroids, padded
        cl[(i >> 5) * CSTR + (i & 31)] = Cb[i];
    stage(0);
    __syncthreads();
    if (t < Kk) {
        float ss = 0.f;
#pragma unroll
        for (int e = 0; e < Ee; ++e) { const float v = cl[t * CSTR + e]; ss += v * v; }
        cn2[t] = ss;   // first read is after the loop's first barrier
    }

    const int w = t >> 5, lane = t & 31, half = lane >> 4, l15 = lane & 15;
    float num = 0.f, den = 0.f;

    for (int c = 0; c < NCH; ++c) {
        if (c + 1 < NCH) { stage(c + 1); wait_async_6(); } else { wait_async_0(); }
        __syncthreads();
        const float* mt = Mt[c & 1];
        const float* et = Et[c & 1];
        if (t < CH) {
            float ss = 0.f;
#pragma unroll
            for (int e = 0; e < Ee; ++e) { const float v = et[t * ESTR2 + e]; ss += v * v; }
            en2[t] = ss;
        }
        __syncthreads();

        for (int ti = w; ti < 16; ti += 8) {           // 2 tiles / wave, uniform
            const int nb = (ti >> 2) * 16, kb = (ti & 3) * 16;
            v8f acc = {};
#pragma unroll
            for (int e0 = 0; e0 < Ee; e0 += 4) {
                const int ec = e0 + 2 * half;
                v2f a, bb;
                a[0]  = et[(nb + l15) * ESTR2 + ec];       // A: M=n rows, wmma-K=e
                a[1]  = et[(nb + l15) * ESTR2 + ec + 1];
                bb[0] = cl[(kb + l15) * CSTR + ec];        // B: cols=k, rows=e
                bb[1] = cl[(kb + l15) * CSTR + ec + 1];
                acc = __builtin_amdgcn_wmma_f32_16x16x4_f32(
                          false, a, false, bb, (short)0, acc, false, false);
            }
#pragma unroll
            for (int r = 0; r < 8; ++r) {
                const int nl = nb + r + 8 * half;
                const int k  = kb + l15;
                const float d2   = en2[nl] + cn2[k] - 2.f * acc[r];
                const float dist = sqrtf(fmaxf(d2, 0.f));
                const float h    = fmaxf(dist - DELTA_V, 0.f);
                const float mval = mt[nl * MSTR + k] * ((k < sz) ? 1.f : 0.f);
                num += h * h * mval;
                den += mval;
            }
        }
        __syncthreads();
    }

    red[t] = num; red[256 + t] = den;
    __syncthreads();
    for (int st = 128; st > 0; st >>= 1) {
        if (t < st) { red[t] += red[t + st]; red[256 + t] += red[256 + t + st]; }
        __syncthreads();
    }
    if (t == 0) { pnum[blockIdx.x] = red[0]; pden[blockIdx.x] = red[256]; }
}

// ---------------------------------------------------------------------------
// Kernel 3: L_d (pairwise centroid hinges) and L_r (centroid norms). Tiny.
// ---------------------------------------------------------------------------
__global__ __launch_bounds__(64) void k3_ldlr(
    const int* __restrict__ size, const float* __restrict__ cent,
    float* __restrict__ ld, float* __restrict__ lr)
{
    const int b = blockIdx.x;
    const int k = threadIdx.x;
    const int sz = size[b];
    __shared__ float cl[Kk * Ee];
    __shared__ float rh[Kk], rc[Kk];
    const float* Cb = cent + (size_t)b * Kk * Ee;
#pragma unroll
    for (int e = 0; e < Ee; ++e) cl[k * Ee + e] = Cb[k * Ee + e];
    __syncthreads();

    const bool vk = k < sz;
    float cn2 = 0.f;
#pragma unroll
    for (int e = 0; e < Ee; ++e) { const float v = cl[k * Ee + e]; cn2 += v * v; }
    const float cnorm = vk ? sqrtf(cn2) : 0.f;

    float hs = 0.f;
    if (vk) {
        for (int j = 0; j < Kk; ++j) {
            if (j == k || j >= sz) continue;
            float d2 = 0.f;
#pragma unroll
            for (int e = 0; e < Ee; ++e) {
                const float d = cl[k * Ee + e] - cl[j * Ee + e];
                d2 += d * d;
            }
            const float h = fmaxf(2.f * DELTA_D - sqrtf(d2), 0.f);
            hs += h * h;
        }
    }
    rh[k] = hs; rc[k] = cnorm;
    __syncthreads();
    if (k == 0) {
        float hsum = 0.f, csum = 0.f;
        for (int j = 0; j < Kk; ++j) { hsum += rh[j]; csum += rc[j]; }
        const float nf = (float)sz;
        ld[b] = (sz > 1) ? hsum / fmaxf(nf * (nf - 1.f), 1.f) : 0.f;
        lr[b] = csum / nf;
    }
}

// ---------------------------------------------------------------------------
// Kernel 4: combine slice partials + terms into the scalar loss.
// ---------------------------------------------------------------------------
__global__ void k4_final(const float* __restrict__ pnum, const float* __restrict__ pden,
                         const float* __restrict__ ld, const float* __restrict__ lr,
                         float* __restrict__ out)
{
    if (threadIdx.x == 0 && blockIdx.x == 0) {
        float av = 0.f, ad = 0.f, ar = 0.f;
        for (int b = 0; b < Bsz; ++b) {
            float n = 0.f, d = 0.f;
            for (int s = 0; s < S2; ++s) { n += pnum[b * S2 + s]; d += pden[b * S2 + s]; }
            av += n / d;
            ad += ld[b];
            ar += lr[b];
        }
        out[0] = (ALPHA * av + BETA * ad + GAMMA * ar) * (1.f / (float)Bsz);
    }
}

extern "C" void kernel_launch(void* const* d_in, const int* in_sizes, int n_in,
                              void* d_out, int out_size, void* d_ws, size_t ws_size,
                              hipStream_t stream)
{
    (void)in_sizes; (void)n_in; (void)out_size; (void)ws_size;
    const float* emb   = (const float*)d_in[0];   // [B,N,E] f32
    const float* masks = (const float*)d_in[1];   // [B,N,K] f32
    const int*   size  = (const int*)d_in[2];     // [B] i32
    float* out = (float*)d_out;

    float* ws   = (float*)d_ws;
    float* cent = ws;                                  // 32768
    float* psum = cent + (size_t)Bsz * Kk * Ee;        // B*S1*K*E = 262144
    float* pcnt = psum + (size_t)Bsz * S1 * Kk * Ee;   // B*S1*K   = 8192
    float* pnum = pcnt + (size_t)Bsz * S1 * Kk;        // B*S2     = 256
    float* pden = pnum + (size_t)Bsz * S2;             // 256
    float* ld   = pden + (size_t)Bsz * S2;             // 16
    float* lr   = ld + Bsz;                            // 16

    k1_partial <<<Bsz * S1, 256, 0, stream>>>(emb, masks, psum, pcnt);
    k1b_combine<<<Bsz,      256, 0, stream>>>(psum, pcnt, size, cent);
    k2_partial <<<Bsz * S2, 256, 0, stream>>>(emb, masks, size, cent, pnum, pden);
    k3_ldlr    <<<Bsz,      64,  0, stream>>>(size, cent, ld, lr);
    k4_final   <<<1,        32,  0, stream>>>(pnum, pden, ld, lr, out);
}